// DynamicFNO1d_51376398795174
// MI455X (gfx1250) — compile-verified
//
#include <hip/hip_runtime.h>
#include <math.h>

typedef __attribute__((ext_vector_type(2))) float v2f;
typedef __attribute__((ext_vector_type(8))) float v8f;

#define F_ACCUM 1
#define F_GELU  2

static __device__ __forceinline__ v8f wmma_f32(v2f a, v2f b, v8f c) {
    return __builtin_amdgcn_wmma_f32_16x16x4_f32(false, a, false, b, (short)0, c, false, false);
}

static __device__ __forceinline__ unsigned lds_off(const void* p) {
    // Generic LDS addresses carry the group-segment byte offset in the low 32 bits.
    return (unsigned)(unsigned long long)p;
}

static __device__ __forceinline__ void async_copy_b32(unsigned lds, const float* g) {
    // VGLOBAL: GLOBAL_LOAD_ASYNC_TO_LDS_B32  vdst(LDS addr), vaddr(64b global), saddr=off
    asm volatile("global_load_async_to_lds_b32 %0, %1, off"
                 :: "v"(lds), "v"((unsigned long long)g) : "memory");
}

static __device__ __forceinline__ void wait_async0() {
    asm volatile("s_wait_asynccnt 0x0" ::: "memory");
}

// ---------------------------------------------------------------------------
// Generic strided / batched f32 WMMA GEMM:  C[b] (op) A[b][M,K] * B[b][K,N]
// Workgroup tile 128x64 (8 waves, each 32x32), K staged 16 at a time through a
// double-buffered LDS tile filled by async global->LDS copies (ASYNCcnt).
// Requires M%128==0, N%64==0, K%16==0 (all call sites satisfy this).
// biasMode: 0 none, 1 per-row, 2 per-col.  flags: F_ACCUM, F_GELU.
// ---------------------------------------------------------------------------
__global__ void __launch_bounds__(256)
gemm_wmma_f32(const float* __restrict__ A, long long aB, int aRS, int aCS,
              const float* __restrict__ Bm, long long bB, int bRS, int bCS,
              float* __restrict__ C, long long cB, int cRS, int cCS,
              int M, int N, int K,
              const float* __restrict__ bias, int biasMode, int flags)
{
    __shared__ float As[2][128][17];
    __shared__ float Bs[2][16][65];

    const int bz = blockIdx.z;
    A  += (long long)bz * aB;
    Bm += (long long)bz * bB;
    C  += (long long)bz * cB;

    const int tid  = threadIdx.x;
    const int wid  = tid >> 5;
    const int lane = tid & 31;
    const int lm   = lane & 15;
    const int kh   = (lane >> 4) << 1;          // K half select: 0 or 2
    const int mo   = (wid >> 1) << 5;           // 0,32,64,96
    const int no   = (wid & 1) << 5;            // 0,32
    const int m0   = blockIdx.x << 7;
    const int n0   = blockIdx.y << 6;

    // Async-stage one 128x16 A tile + 16x64 B tile into LDS buffer `buf`.
    auto issue_stage = [&](int k0, int buf) {
        for (int i = tid; i < 128 * 16; i += 256) {
            int mm = i >> 4, kk = i & 15;
            const float* g = &A[(long long)(m0 + mm) * aRS + (long long)(k0 + kk) * aCS];
            async_copy_b32(lds_off(&As[buf][mm][kk]), g);
        }
        for (int i = tid; i < 16 * 64; i += 256) {
            int kk = i >> 6, nn = i & 63;
            const float* g = &Bm[(long long)(k0 + kk) * bRS + (long long)(n0 + nn) * bCS];
            async_copy_b32(lds_off(&Bs[buf][kk][nn]), g);
        }
    };

    v8f acc00 = {}, acc01 = {}, acc10 = {}, acc11 = {};

    const int S = K >> 4;
    issue_stage(0, 0);
    for (int s = 0; s < S; ++s) {
        wait_async0();          // this wave's copies for stage s are in LDS
        __syncthreads();        // all waves' copies visible; stage s-1 reads done
        if (s + 1 < S) issue_stage((s + 1) << 4, (s + 1) & 1);  // overlaps compute
        const int buf = s & 1;
#pragma unroll
        for (int kk = 0; kk < 16; kk += 4) {
            v2f a0 = { As[buf][mo + lm][kk + kh],      As[buf][mo + lm][kk + kh + 1] };
            v2f a1 = { As[buf][mo + 16 + lm][kk + kh], As[buf][mo + 16 + lm][kk + kh + 1] };
            v2f b0 = { Bs[buf][kk + kh][no + lm],      Bs[buf][kk + kh + 1][no + lm] };
            v2f b1 = { Bs[buf][kk + kh][no + 16 + lm], Bs[buf][kk + kh + 1][no + 16 + lm] };
            acc00 = wmma_f32(a0, b0, acc00);
            acc01 = wmma_f32(a0, b1, acc01);
            acc10 = wmma_f32(a1, b0, acc10);
            acc11 = wmma_f32(a1, b1, acc11);
        }
    }

    const int rhalf = (lane >> 4) << 3;
#pragma unroll
    for (int ms = 0; ms < 2; ++ms) {
        v8f am0 = ms ? acc10 : acc00;
        v8f am1 = ms ? acc11 : acc01;
        const int rbase = m0 + mo + (ms << 4) + rhalf;
        const int c0 = n0 + no + lm;
        const int c1 = c0 + 16;
#pragma unroll
        for (int r = 0; r < 8; ++r) {
            int row = rbase + r;
            long long off0 = (long long)row * cRS + (long long)c0 * cCS;
            long long off1 = (long long)row * cRS + (long long)c1 * cCS;
            float v0 = am0[r], v1 = am1[r];
            if (flags & F_ACCUM) { v0 += C[off0]; v1 += C[off1]; }
            if (biasMode == 1)   { v0 += bias[row]; v1 += bias[row]; }
            else if (biasMode == 2) { v0 += bias[c0]; v1 += bias[c1]; }
            if (flags & F_GELU) {
                v0 = 0.5f * v0 * (1.0f + erff(v0 * 0.70710678118654752f));
                v1 = 0.5f * v1 * (1.0f + erff(v1 * 0.70710678118654752f));
            }
            C[off0] = v0;
            C[off1] = v1;
        }
    }
}

// ---------------------------------------------------------------------------
// Batched complex mix: out[b,o,f(re|im)] = sum_i Wg[b,i,o] * hf[b,i,f]
// Wg given as separate re/im [B,128,128]; hf/out rows = (b,ch), 256 cols re|im.
// ---------------------------------------------------------------------------
__global__ void __launch_bounds__(256)
cgemm_mix(const float* __restrict__ WgR, const float* __restrict__ WgI,
          const float* __restrict__ hf, float* __restrict__ outm)
{
    const int b    = blockIdx.y;
    const int wid  = threadIdx.x >> 5;
    const int lane = threadIdx.x & 31;
    const int t    = blockIdx.x * 8 + wid;      // 0..63 tiles
    const int o0   = (t >> 3) << 4;
    const int f0   = (t & 7) << 4;
    const int lm   = lane & 15;
    const int kh   = (lane >> 4) << 1;

    const float* wr = WgR + (long long)b * 16384;
    const float* wi = WgI + (long long)b * 16384;
    const float* hb = hf  + (long long)b * 128 * 256;

    v8f cre = {};
    v8f cim = {};
    for (int i0 = 0; i0 < 128; i0 += 4) {
        int k = i0 + kh;
        int o = o0 + lm;
        v2f ar  = { wr[k * 128 + o], wr[(k + 1) * 128 + o] };
        v2f ai  = { wi[k * 128 + o], wi[(k + 1) * 128 + o] };
        v2f nai = { -ai.x, -ai.y };
        v2f br  = { hb[k * 256 + f0 + lm],       hb[(k + 1) * 256 + f0 + lm] };
        v2f bi  = { hb[k * 256 + 128 + f0 + lm], hb[(k + 1) * 256 + 128 + f0 + lm] };
        cre = wmma_f32(ar,  br, cre);
        cre = wmma_f32(nai, bi, cre);
        cim = wmma_f32(ai,  br, cim);
        cim = wmma_f32(ar,  bi, cim);
    }
    float* ob = outm + (long long)b * 128 * 256;
    const int rb = o0 + ((lane >> 4) << 3);
#pragma unroll
    for (int r = 0; r < 8; ++r) {
        int row = rb + r;
        ob[row * 256 + f0 + lm]       = cre[r];
        ob[row * 256 + 128 + f0 + lm] = cim[r];
    }
}

// ---------------------------------------------------------------------------
// Table kernels (recomputed each call; deterministic)
// ---------------------------------------------------------------------------
__global__ void k_dft_tables(float* __restrict__ dftB) {
    int idx = blockIdx.x * 256 + threadIdx.x;          // 4096*256
    if (idx >= 4096 * 256) return;
    int n = idx >> 8, j = idx & 255, f = j & 127;
    int m = (f * n) & 4095;                            // exact phase reduction
    double th = (double)m * (6.283185307179586476925287 / 4096.0);
    dftB[idx] = (j < 128) ? (float)cos(th) : (float)(-sin(th));
}

__global__ void k_idft_tables(float* __restrict__ idftB) {
    int idx = blockIdx.x * 256 + threadIdx.x;          // 256*4096
    if (idx >= 256 * 4096) return;
    int fp = idx >> 12, n = idx & 4095, f = fp & 127;
    int m = (f * n) & 4095;
    double th = (double)m * (6.283185307179586476925287 / 4096.0);
    float v;
    if (fp < 128) v = (float)(((f == 0) ? 1.0 : 2.0) / 4096.0 * cos(th));
    else          v = (float)(-2.0 / 4096.0 * sin(th));
    idftB[idx] = v;
}

__global__ void k_cft_tables(float* __restrict__ Wc, int* __restrict__ li,
                             int* __restrict__ ri, float* __restrict__ wl,
                             float* __restrict__ wr) {
    const double PI = 3.14159265358979323846;
    int tid = threadIdx.x;
    for (int p = tid; p < 160; p += 256) {
        int l = p >> 3, m = p & 7;
        double node = -cos((2.0 * m + 1.0) * PI / 16.0);
        double v = l / 20.0 + 0.025 * (node + 1.0);
        int r = (int)ceil(v * 4095.0);
        if (r > 4095) r = 4095;
        if (r < 0) r = 0;
        int lf = r - 1; if (lf < 0) lf = 0;
        double tl = lf / 4095.0, tr = r / 4095.0;
        double den = (tr == tl) ? 1.0 : (tr - tl);
        double wrv = (v - tl) / den;
        li[p] = lf; ri[p] = r;
        wr[p] = (float)wrv; wl[p] = (float)(1.0 - wrv);
    }
    for (int idx = tid; idx < 160 * 128; idx += 256) {
        int p = idx >> 7, f = idx & 127;
        int l = p >> 3, k = p & 7;
        double wp = f * (PI * 0.05);                   // f * 2*pi*seg/2
        double cr = 0.0, ci = 0.0;
        for (int m = 0; m < 8; ++m) {
            double node = -cos((2.0 * m + 1.0) * PI / 16.0);
            double T = cos((double)k * acos(node));
            double q = node * wp;
            cr += T * cos(q);
            ci += T * (-sin(q));
        }
        cr *= 0.025; ci *= 0.025;
        int ph = (f * l) % 20;
        double th = 2.0 * PI * (double)ph / 20.0;
        double sr = cos(th), si = -sin(th);
        Wc[p * 256 + f]       = (float)(sr * cr - si * ci);
        Wc[p * 256 + 128 + f] = (float)(sr * ci + si * cr);
    }
}

// ---------------------------------------------------------------------------
// Small elementwise / reduce kernels
// ---------------------------------------------------------------------------
__global__ void k_fc0(const float* __restrict__ x, const float* __restrict__ fc0W,
                      const float* __restrict__ fc0b, float* __restrict__ h) {
    long long idx = (long long)blockIdx.x * 256 + threadIdx.x;  // 64*128*4096
    if (idx >= (long long)64 * 128 * 4096) return;
    int n = (int)(idx & 4095);
    int w = (int)((idx >> 12) & 127);
    int b = (int)(idx >> 19);
    float g = (float)n * (1.0f / 4095.0f);
    h[idx] = fc0W[2 * w] * x[(long long)b * 4096 + n] + fc0W[2 * w + 1] * g + fc0b[w];
}

__global__ void k_sig(const float* __restrict__ h, const int* __restrict__ li,
                      const int* __restrict__ ri, const float* __restrict__ wl,
                      const float* __restrict__ wr, float* __restrict__ sig) {
    long long idx = (long long)blockIdx.x * 256 + threadIdx.x;  // 8192*160
    if (idx >= (long long)8192 * 160) return;
    int p = (int)(idx % 160);
    long long row = idx / 160;
    const float* hr = h + row * 4096;
    sig[idx] = wl[p] * hr[li[p]] + wr[p] * hr[ri[p]];
}

__global__ void k_mag(const float* __restrict__ spec, float* __restrict__ mag) {
    int idx = blockIdx.x * 256 + threadIdx.x;                  // 64*128
    if (idx >= 8192) return;
    int b = idx >> 7, f = idx & 127;
    float s = 0.0f;
    for (int c = 0; c < 128; ++c) {
        const float* r = spec + ((long long)(b * 128 + c)) * 256;
        float re = r[f], im = r[128 + f];
        s += sqrtf(re * re + im * im);
    }
    mag[idx] = s * (1.0f / 128.0f);
}

__global__ void k_gate(const float* __restrict__ mag, const float* __restrict__ gW,
                       const float* __restrict__ gb, float* __restrict__ gate) {
    int idx = blockIdx.x * 256 + threadIdx.x;                  // 64*128
    if (idx >= 8192) return;
    int b = idx >> 7, w = idx & 127;
    float s = gb[w];
    const float* mr = mag + b * 128;
    const float* wrow = gW + w * 128;
    for (int f = 0; f < 128; ++f) s += mr[f] * wrow[f];
    gate[idx] = 1.0f / (1.0f + expf(-s));
}

__global__ void k_wsum(const float* __restrict__ wAre, const float* __restrict__ wAim,
                       const float* __restrict__ wBre, const float* __restrict__ wBim,
                       float* __restrict__ ar, float* __restrict__ ai,
                       float* __restrict__ br, float* __restrict__ bi) {
    int io = blockIdx.x * 256 + threadIdx.x;                   // 128*128
    if (io >= 16384) return;
    long long base = (long long)io * 128;
    float s0 = 0, s1 = 0, s2 = 0, s3 = 0;
    for (int y = 0; y < 128; ++y) {
        s0 += wAre[base + y]; s1 += wAim[base + y];
        s2 += wBre[base + y]; s3 += wBim[base + y];
    }
    ar[io] = s0; ai[io] = s1; br[io] = s2; bi[io] = s3;
}

// NOTE: gate[:,None,:]*wA[None] broadcasts gate over the OUT axis -> g = gate[b,o]
__global__ void k_wg(const float* __restrict__ gate,
                     const float* __restrict__ ar, const float* __restrict__ ai,
                     const float* __restrict__ br, const float* __restrict__ bi,
                     float* __restrict__ WgR, float* __restrict__ WgI) {
    long long idx = (long long)blockIdx.x * 256 + threadIdx.x; // 64*16384
    if (idx >= (long long)64 * 16384) return;
    int io = (int)(idx & 16383);
    int b  = (int)(idx >> 14);
    int o  = io & 127;
    float g = gate[(b << 7) + o];
    WgR[idx] = g * ar[io] + (1.0f - g) * br[io];
    WgI[idx] = g * ai[io] + (1.0f - g) * bi[io];
}

__global__ void k_fc2(const float* __restrict__ t, const float* __restrict__ w2,
                      const float* __restrict__ b2, float* __restrict__ out) {
    int idx = blockIdx.x * 256 + threadIdx.x;                  // 64*4096
    if (idx >= 262144) return;
    const float* r = t + (long long)idx * 128;
    float s = b2[0];
    for (int j = 0; j < 128; ++j) s += r[j] * w2[j];
    out[idx] = s;
}

// ---------------------------------------------------------------------------
extern "C" void kernel_launch(void* const* d_in, const int* in_sizes, int n_in,
                              void* d_out, int out_size, void* d_ws, size_t ws_size,
                              hipStream_t stream) {
    (void)in_sizes; (void)n_in; (void)out_size; (void)ws_size;
    const float* x     = (const float*)d_in[0];
    const float* fc0W  = (const float*)d_in[1];
    const float* fc0b  = (const float*)d_in[2];
    const float* gateW = (const float*)d_in[3];
    const float* gateb = (const float*)d_in[4];
    const float* wAre  = (const float*)d_in[5];
    const float* wAim  = (const float*)d_in[6];
    const float* wBre  = (const float*)d_in[7];
    const float* wBim  = (const float*)d_in[8];
    const float* convW = (const float*)d_in[9];
    const float* convb = (const float*)d_in[10];
    const float* fc1W  = (const float*)d_in[11];
    const float* fc1b  = (const float*)d_in[12];
    const float* fc2W  = (const float*)d_in[13];
    const float* fc2b  = (const float*)d_in[14];

    float* ws = (float*)d_ws;
    size_t o = 0;
    float* H0   = ws + o; o += (size_t)33554432;   // [B,W,N]
    float* H1   = ws + o; o += (size_t)33554432;
    float* HF   = ws + o; o += (size_t)2097152;    // [8192,256] re|im
    float* OUTM = ws + o; o += (size_t)2097152;
    float* SPEC = ws + o; o += (size_t)2097152;
    float* SIG  = ws + o; o += (size_t)1310720;    // [8192,160]
    float* DFTB = ws + o; o += (size_t)1048576;    // [4096,256]
    float* IDFT = ws + o; o += (size_t)1048576;    // [256,4096]
    float* WC   = ws + o; o += (size_t)40960;      // [160,256]
    int*   LI   = (int*)(ws + o); o += 256;
    int*   RI   = (int*)(ws + o); o += 256;
    float* WL   = ws + o; o += 256;
    float* WR   = ws + o; o += 256;
    float* MAG  = ws + o; o += 8192;
    float* GATE = ws + o; o += 8192;
    float* WASR = ws + o; o += 16384;
    float* WASI = ws + o; o += 16384;
    float* WBSR = ws + o; o += 16384;
    float* WBSI = ws + o; o += 16384;
    float* WGR  = ws + o; o += (size_t)1048576;    // [B,128,128]
    float* WGI  = ws + o; o += (size_t)1048576;

    // constant tables
    k_dft_tables<<<4096, 256, 0, stream>>>(DFTB);
    k_idft_tables<<<4096, 256, 0, stream>>>(IDFT);
    k_cft_tables<<<1, 256, 0, stream>>>(WC, LI, RI, WL, WR);

    // fc0 + grid concat -> h [B,W,N]
    k_fc0<<<131072, 256, 0, stream>>>(x, fc0W, fc0b, H0);

    float* hc = H0;
    float* hn = H1;
    for (int l = 0; l < 4; ++l) {
        // ---- CFT magnitude -> gate ----
        k_sig<<<5120, 256, 0, stream>>>(hc, LI, RI, WL, WR, SIG);
        gemm_wmma_f32<<<dim3(64, 4, 1), 256, 0, stream>>>(
            SIG, 0, 160, 1,  WC, 0, 256, 1,  SPEC, 0, 256, 1,
            8192, 256, 160, nullptr, 0, 0);
        k_mag<<<32, 256, 0, stream>>>(SPEC, MAG);
        k_gate<<<32, 256, 0, stream>>>(MAG, gateW + (size_t)l * 16384,
                                       gateb + (size_t)l * 128, GATE);
        // ---- gated spectral weights ----
        k_wsum<<<64, 256, 0, stream>>>(wAre + (size_t)l * 2097152,
                                       wAim + (size_t)l * 2097152,
                                       wBre + (size_t)l * 2097152,
                                       wBim + (size_t)l * 2097152,
                                       WASR, WASI, WBSR, WBSI);
        k_wg<<<4096, 256, 0, stream>>>(GATE, WASR, WASI, WBSR, WBSI, WGR, WGI);
        // ---- forward DFT (first 128 modes) ----
        gemm_wmma_f32<<<dim3(64, 4, 1), 256, 0, stream>>>(
            hc, 0, 4096, 1,  DFTB, 0, 256, 1,  HF, 0, 256, 1,
            8192, 256, 4096, nullptr, 0, 0);
        // ---- complex spectral mix ----
        cgemm_mix<<<dim3(8, 64), 256, 0, stream>>>(WGR, WGI, HF, OUTM);
        // ---- pointwise conv x2 (+bias) into hn ----
        gemm_wmma_f32<<<dim3(1, 64, 64), 256, 0, stream>>>(
            convW + (size_t)l * 16384, 0, 128, 1,
            hc, 524288, 4096, 1,
            hn, 524288, 4096, 1,
            128, 4096, 128, convb + (size_t)l * 128, 1, 0);
        // ---- inverse DFT, accumulate, (gelu if l<3) ----
        gemm_wmma_f32<<<dim3(64, 64, 1), 256, 0, stream>>>(
            OUTM, 0, 256, 1,  IDFT, 0, 4096, 1,  hn, 0, 4096, 1,
            8192, 4096, 256, nullptr, 0, F_ACCUM | ((l < 3) ? F_GELU : 0));
        float* tmp = hc; hc = hn; hn = tmp;
    }

    // fc1 (batched, A = h^T per batch via strides) + bias(col) + gelu -> hn as [B,N,128]
    gemm_wmma_f32<<<dim3(32, 2, 64), 256, 0, stream>>>(
        hc, 524288, 1, 4096,
        fc1W, 0, 1, 128,
        hn, 524288, 128, 1,
        4096, 128, 128, fc1b, 2, F_GELU);

    // fc2 -> d_out [B,N,1]
    k_fc2<<<1024, 256, 0, stream>>>(hn, fc2W, fc2b, (float*)d_out);
}